// downprompt_31628139168248
// MI455X (gfx1250) — compile-verified
//
#include <hip/hip_runtime.h>
#include <math.h>
#include <stdint.h>

typedef __attribute__((ext_vector_type(2))) float v2f;
typedef __attribute__((ext_vector_type(8))) float v8f;
typedef int vi4 __attribute__((vector_size(16)));            // matches builtin param pointee
typedef __attribute__((address_space(1))) vi4 gvi4;          // global (AS1)
typedef __attribute__((address_space(3))) vi4 lvi4;          // LDS (AS3)

#define DDIM 256
#define NCLS 7
#define CEPS 1e-8f
#define A4C  0.1f

#define BLOCK_THREADS 128           // 4 waves
#define ROWS_PER_BLOCK 64           // 16 rows per wave tile
#define KCHUNK 32
#define NCHUNK (DDIM / KCHUNK)      // 8
#define RSTRIDE 36                  // 32 + 4 pad: 16B-aligned b128, conflict-free b64 reads
#define BSTRIDE 258                 // 256 + 2 pad for aveT rows

#if __has_builtin(__builtin_amdgcn_global_load_async_to_lds_b128)
#define HAVE_ASYNC_LDS 1
#else
#define HAVE_ASYNC_LDS 0
#endif

#if __has_builtin(__builtin_amdgcn_s_wait_asynccnt)
#define WAIT_ASYNC(n) __builtin_amdgcn_s_wait_asynccnt(n)
#else
#define WAIT_ASYNC(n) asm volatile("s_wait_asynccnt %0" ::"i"(n) : "memory")
#endif

__device__ __forceinline__ float elu1(float x) {
    return (x > 0.f) ? x : (__expf(x) - 1.f);
}

// 16B global->LDS copy: async (bypasses VGPRs, tracked by ASYNCcnt) when available.
__device__ __forceinline__ void cp16_g2l(const float* gp, float* lp) {
#if HAVE_ASYNC_LDS
    __builtin_amdgcn_global_load_async_to_lds_b128(
        (gvi4*)(uintptr_t)gp,
        (lvi4*)(uintptr_t)lp,
        0, 0);
#else
    *(float4*)lp = *(const float4*)gp;
#endif
}

__device__ __forceinline__ void issue_chunk(const float* __restrict__ seq,
                                            const float* __restrict__ seq1,
                                            float* sb, float* s1b,
                                            int blockRow0, int kc, int t, int N) {
    #pragma unroll
    for (int p = 0; p < 4; ++p) {
        int linear = p * (BLOCK_THREADS * 4) + t * 4;   // float index in 64x32 chunk
        int r    = linear >> 5;                          // 0..63
        int cofs = linear & 31;                          // 0,4,...,28
        int grow = blockRow0 + r;
        grow = (grow < N) ? grow : (N - 1);              // clamp: uniform EXEC, no garbage LDS
        size_t goff = (size_t)grow * DDIM + kc + cofs;
        int lofs = r * RSTRIDE + cofs;
        cp16_g2l(seq + goff, sb + lofs);
        cp16_g2l(seq1 + goff, s1b + lofs);
    }
}

// ws layout: ws[0..255] = g[d], ws[256..271] = padded prototype norms
__global__ void downprompt_prep(const float* __restrict__ prompt,
                                const float* __restrict__ w_np,
                                const float* __restrict__ w_ds,
                                const float* __restrict__ w_df,
                                const float* __restrict__ ave,
                                float* __restrict__ ws) {
    int t = threadIdx.x;           // 256 threads
    float a = w_df[0], b = w_df[1];
    float w = w_np[0] * prompt[t] + w_np[1] * prompt[DDIM + t] + w_np[2] * prompt[2 * DDIM + t];
    ws[t] = a * (1.f + elu1(w)) + b * w_ds[t];
    if (t < 16) {
        float na = 0.f;
        if (t < NCLS) {
            float s = 0.f;
            for (int d = 0; d < DDIM; ++d) { float v = ave[t * DDIM + d]; s += v * v; }
            na = sqrtf(s);
        }
        ws[DDIM + t] = na;
    }
}

__global__ __launch_bounds__(BLOCK_THREADS) void
downprompt_main(const float* __restrict__ seq,
                const float* __restrict__ seq1,
                const float* __restrict__ ave,
                const float* __restrict__ ws,
                float* __restrict__ out,
                int N) {
    __shared__ float gsm[DDIM];
    __shared__ float nasm[16];
    __shared__ float bsm[16 * BSTRIDE];                     // aveT, zero-padded to 16 cols
    __shared__ float sbuf[2][ROWS_PER_BLOCK * RSTRIDE];     // raw seq chunks (double buffer)
    __shared__ float s1buf[2][ROWS_PER_BLOCK * RSTRIDE];    // raw seq1 chunks

    const int t    = threadIdx.x;
    const int lane = t & 31;
    const int wave = t >> 5;
    const int n    = lane & 15;      // class column / A-row index
    const int h    = lane >> 4;      // half-wave select (K pair 2h, 2h+1)
    const int blockRow0 = blockIdx.x * ROWS_PER_BLOCK;
    const int tileRow0  = blockRow0 + wave * 16;

    gsm[t] = ws[t];
    gsm[t + 128] = ws[t + 128];
    if (t < 16) nasm[t] = ws[DDIM + t];
    #pragma unroll
    for (int nn = 0; nn < 16; ++nn) {
        bsm[nn * BSTRIDE + t]       = (nn < NCLS) ? ave[nn * DDIM + t] : 0.f;
        bsm[nn * BSTRIDE + t + 128] = (nn < NCLS) ? ave[nn * DDIM + t + 128] : 0.f;
    }

    // prologue: stream chunk 0
    issue_chunk(seq, seq1, sbuf[0], s1buf[0], blockRow0, 0, t, N);

    v8f acc = {};
    float sumsq = 0.f;
    const int rowofs = (wave * 16 + n) * RSTRIDE;

    for (int kci = 0; kci < NCHUNK; ++kci) {
        const int cur = kci & 1;
        const int kc  = kci * KCHUNK;

        if (kci + 1 < NCHUNK) {
            // stream chunk k+1 while chunk k is consumed below
            issue_chunk(seq, seq1, sbuf[1 - cur], s1buf[1 - cur],
                        blockRow0, kc + KCHUNK, t, N);
            WAIT_ASYNC(8);          // batch k done (in-order), batch k+1 may remain
        } else {
            WAIT_ASYNC(0);
        }
        __syncthreads();            // all waves' chunk-k data visible (also covers LDS fills)

        const float* sb  = sbuf[cur];
        const float* s1b = s1buf[cur];
        #pragma unroll
        for (int kk = 0; kk < KCHUNK; kk += 4) {
            int kl = kk + 2 * h;
            v2f s  = *(const v2f*)&sb[rowofs + kl];
            v2f s1 = *(const v2f*)&s1b[rowofs + kl];
            v2f gg = *(const v2f*)&gsm[kc + kl];
            v2f aF;
            float x0 = gg.x * s.x, x1 = gg.y * s.y;
            aF.x = ((x0 > 0.f) ? x0 : (__expf(x0) - 1.f)) + A4C * s1.x;
            aF.y = ((x1 > 0.f) ? x1 : (__expf(x1) - 1.f)) + A4C * s1.y;
            sumsq += aF.x * aF.x + aF.y * aF.y;
            v2f bF = *(const v2f*)&bsm[n * BSTRIDE + kc + kl];
            acc = __builtin_amdgcn_wmma_f32_16x16x4_f32(
                false, aF, false, bF, (short)0, acc, false, false);
        }
        __syncthreads();            // everyone done reading buf[cur] before it is overwritten
    }

    // ---- norms: lanes r and r+16 jointly hold row r -> one xor-16 add ----
    sumsq += __shfl_xor(sumsq, 16);
    float rowNorm = sqrtf(sumsq);        // lane L holds norm of tile-row (L&15)
    float nav = nasm[n];

    // ---- cosine + softmax per D-matrix VGPR (row m = j + 8h, class = n) ----
    #pragma unroll
    for (int j = 0; j < 8; ++j) {
        int m   = j + 8 * h;
        int row = tileRow0 + m;
        float nr    = __shfl(rowNorm, m);            // bpermute: per-lane src ok
        float denom = fmaxf(nr * nav, CEPS);
        float sim   = (n < NCLS) ? acc[j] / denom : -INFINITY;
        float mx = sim;
        mx = fmaxf(mx, __shfl_xor(mx, 1));
        mx = fmaxf(mx, __shfl_xor(mx, 2));
        mx = fmaxf(mx, __shfl_xor(mx, 4));
        mx = fmaxf(mx, __shfl_xor(mx, 8));
        float pv = (n < NCLS) ? __expf(sim - mx) : 0.f;
        float sm = pv;
        sm += __shfl_xor(sm, 1);
        sm += __shfl_xor(sm, 2);
        sm += __shfl_xor(sm, 4);
        sm += __shfl_xor(sm, 8);
        if (n < NCLS && row < N)
            out[(size_t)row * NCLS + n] = pv / sm;
    }
}

extern "C" void kernel_launch(void* const* d_in, const int* in_sizes, int n_in,
                              void* d_out, int out_size, void* d_ws, size_t ws_size,
                              hipStream_t stream) {
    const float* seq    = (const float*)d_in[0];
    const float* seq1   = (const float*)d_in[1];
    const float* prompt = (const float*)d_in[2];
    const float* w_np   = (const float*)d_in[3];
    const float* w_ds   = (const float*)d_in[4];
    const float* w_df   = (const float*)d_in[5];
    const float* ave    = (const float*)d_in[6];
    float* out = (float*)d_out;
    float* ws  = (float*)d_ws;

    const int N = in_sizes[0] / DDIM;

    downprompt_prep<<<1, 256, 0, stream>>>(prompt, w_np, w_ds, w_df, ave, ws);

    int blocks = (N + ROWS_PER_BLOCK - 1) / ROWS_PER_BLOCK;
    downprompt_main<<<blocks, BLOCK_THREADS, 0, stream>>>(seq, seq1, ave, ws, out, N);
}